// GraphSAGEWithAttention_34127810134469
// MI455X (gfx1250) — compile-verified
//
#include <hip/hip_runtime.h>

#define NN 100000
#define EE 800000
#define DH 128
#define HHD 4
#define FHH 32
#define NEG_SLOPE 0.2f
#define LN_EPS 1e-5f
#define LDS_PAD 4
#define LDS_W (DH + LDS_PAD)

typedef __attribute__((ext_vector_type(2))) float v2f;
typedef __attribute__((ext_vector_type(8))) float v8f;

// ---------- helpers ----------
__device__ __forceinline__ int fOrd(float f) {
    int i = __float_as_int(f);
    return i >= 0 ? i : (i ^ 0x7FFFFFFF);
}
__device__ __forceinline__ float fUnord(int i) {
    return __int_as_float(i >= 0 ? i : (i ^ 0x7FFFFFFF));
}
__device__ __forceinline__ void edge_sd(int e, const int* __restrict__ src,
                                        const int* __restrict__ dst, int& s, int& d) {
    if (e < EE) { s = src[e]; d = dst[e]; }
    else        { s = e - EE; d = e - EE; }   // self loop
}
__device__ __forceinline__ float eluf(float x) { return x > 0.f ? x : (expf(x) - 1.f); }

// ---------- utility kernels ----------
__global__ void fillf(float* __restrict__ p, float v, int n) {
    int t = blockIdx.x * blockDim.x + threadIdx.x;
    if (t < n) p[t] = v;
}

__global__ void count_deg(const int* __restrict__ dst, float* __restrict__ cnt) {
    int e = blockIdx.x * blockDim.x + threadIdx.x;
    if (e < EE) atomicAdd(&cnt[dst[e]], 1.0f);
}

// SAGE neighbor aggregation: agg[dst] += h[src]; one thread = (edge, 4-float chunk)
__global__ void scatter_add(const float* __restrict__ h, const int* __restrict__ src,
                            const int* __restrict__ dst, float* __restrict__ agg) {
    int t = blockIdx.x * blockDim.x + threadIdx.x;
    if (t >= EE * 32) return;
    int e = t >> 5;
    int c = (t & 31) * 4;
    int s = src[e], d = dst[e];
    const float4 v = *(const float4*)(h + (size_t)s * DH + c);
    float* o = agg + (size_t)d * DH + c;
    atomicAdd(o + 0, v.x);
    atomicAdd(o + 1, v.y);
    atomicAdd(o + 2, v.z);
    atomicAdd(o + 3, v.w);
}

// ---------- fp32 WMMA GEMM (LDS-staged A, 2 row tiles per wave) ----------
// FUSED: out = (A1/max(cnt,1)) @ B1 + A2 @ B2 + bias   (SAGE)
// !FUSED: out = A1 @ B1                                 (GAT projection)
// Block = 256 threads (8 waves): 32 rows x 128 cols. Grid = N/32.
template <bool FUSED>
__global__ void gemm_wmma(const float* __restrict__ A1, const float* __restrict__ cnt,
                          const float* __restrict__ B1, const float* __restrict__ A2,
                          const float* __restrict__ B2, const float* __restrict__ bias,
                          float* __restrict__ out) {
    __shared__ float As[32][LDS_W];
    __shared__ float As2[FUSED ? 32 : 1][FUSED ? LDS_W : 1];

    const int tid  = threadIdx.x;
    const int rowb = blockIdx.x * 32;

    // ---- stage A rows into LDS (scale by 1/deg during staging for FUSED) ----
    {
        const int r  = tid >> 3;          // 0..31
        const int c0 = (tid & 7) * 16;    // 0..112
        const int rg = rowb + r;
        float sc = 1.0f;
        if (FUSED) sc = 1.0f / fmaxf(cnt[rg], 1.0f);
        const float4* pa = (const float4*)(A1 + (size_t)rg * DH + c0);
#pragma unroll
        for (int i = 0; i < 4; ++i) {
            float4 v = pa[i];
            As[r][c0 + 4 * i + 0] = v.x * sc;
            As[r][c0 + 4 * i + 1] = v.y * sc;
            As[r][c0 + 4 * i + 2] = v.z * sc;
            As[r][c0 + 4 * i + 3] = v.w * sc;
        }
        if (FUSED) {
            const float4* pa2 = (const float4*)(A2 + (size_t)rg * DH + c0);
#pragma unroll
            for (int i = 0; i < 4; ++i) {
                float4 v = pa2[i];
                As2[r][c0 + 4 * i + 0] = v.x;
                As2[r][c0 + 4 * i + 1] = v.y;
                As2[r][c0 + 4 * i + 2] = v.z;
                As2[r][c0 + 4 * i + 3] = v.w;
            }
        }
    }
    __syncthreads();

    const int lane = tid & 31;
    const int wave = tid >> 5;
    const int l15  = lane & 15;
    const int hi   = lane >> 4;          // 0: K pair {0,1}, 1: K pair {2,3}
    const int col  = wave * 16 + l15;    // B/C: lane indexes N

    v8f acc0 = {};
    v8f acc1 = {};
    for (int k0 = 0; k0 < DH; k0 += 4) {
        const int ka = k0 + hi * 2;
        v2f b1f, a0, a1;
        b1f[0] = B1[ka * DH + col];
        b1f[1] = B1[(ka + 1) * DH + col];
        a0[0]  = As[l15][ka];
        a0[1]  = As[l15][ka + 1];
        a1[0]  = As[16 + l15][ka];
        a1[1]  = As[16 + l15][ka + 1];
        acc0 = __builtin_amdgcn_wmma_f32_16x16x4_f32(false, a0, false, b1f, (short)0,
                                                     acc0, false, false);
        acc1 = __builtin_amdgcn_wmma_f32_16x16x4_f32(false, a1, false, b1f, (short)0,
                                                     acc1, false, false);
        if (FUSED) {
            v2f b2f, c0f, c1f;
            b2f[0] = B2[ka * DH + col];
            b2f[1] = B2[(ka + 1) * DH + col];
            c0f[0] = As2[l15][ka];
            c0f[1] = As2[l15][ka + 1];
            c1f[0] = As2[16 + l15][ka];
            c1f[1] = As2[16 + l15][ka + 1];
            acc0 = __builtin_amdgcn_wmma_f32_16x16x4_f32(false, c0f, false, b2f, (short)0,
                                                         acc0, false, false);
            acc1 = __builtin_amdgcn_wmma_f32_16x16x4_f32(false, c1f, false, b2f, (short)0,
                                                         acc1, false, false);
        }
    }

    const float bv = FUSED ? bias[col] : 0.0f;
    {
        const int r0 = rowb + hi * 8;    // C layout: vgpr i -> row r0+i
        float* o = out + (size_t)r0 * DH + col;
#pragma unroll
        for (int i = 0; i < 8; ++i) o[(size_t)i * DH] = acc0[i] + bv;
    }
    {
        const int r0 = rowb + 16 + hi * 8;
        float* o = out + (size_t)r0 * DH + col;
#pragma unroll
        for (int i = 0; i < 8; ++i) o[(size_t)i * DH] = acc1[i] + bv;
    }
}

// ---------- LayerNorm + ELU (one wave32 per row, 4 floats/lane) ----------
__global__ void ln_elu(const float* __restrict__ in, const float* __restrict__ g,
                       const float* __restrict__ b, float* __restrict__ out) {
    int wid  = (blockIdx.x * blockDim.x + threadIdx.x) >> 5;
    int lane = threadIdx.x & 31;
    if (wid >= NN) return;
    const float4 v = ((const float4*)(in + (size_t)wid * DH))[lane];
    float s = v.x + v.y + v.z + v.w;
#pragma unroll
    for (int o = 16; o; o >>= 1) s += __shfl_xor(s, o, 32);
    float mu = s * (1.0f / DH);
    float dx = v.x - mu, dy = v.y - mu, dz = v.z - mu, dw = v.w - mu;
    float q = dx * dx + dy * dy + dz * dz + dw * dw;
#pragma unroll
    for (int o = 16; o; o >>= 1) q += __shfl_xor(q, o, 32);
    float rs = rsqrtf(q * (1.0f / DH) + LN_EPS);
    const float4 gg = ((const float4*)g)[lane];
    const float4 bb = ((const float4*)b)[lane];
    float4 r;
    r.x = eluf(dx * rs * gg.x + bb.x);
    r.y = eluf(dy * rs * gg.y + bb.y);
    r.z = eluf(dz * rs * gg.z + bb.z);
    r.w = eluf(dw * rs * gg.w + bb.w);
    ((float4*)(out + (size_t)wid * DH))[lane] = r;
}

// ---------- GAT attention ----------
__global__ void gat_logits(const float* __restrict__ hp, const float* __restrict__ asrc,
                           const float* __restrict__ adst, float* __restrict__ als,
                           float* __restrict__ ald) {
    int t = blockIdx.x * blockDim.x + threadIdx.x;
    if (t >= NN * HHD) return;
    int n = t >> 2, hd = t & 3;
    const float* hpn = hp + (size_t)n * DH + hd * FHH;
    const float* as_ = asrc + hd * FHH;
    const float* ad_ = adst + hd * FHH;
    float s1 = 0.f, s2 = 0.f;
#pragma unroll 8
    for (int f = 0; f < FHH; ++f) {
        float h = hpn[f];
        s1 += h * as_[f];
        s2 += h * ad_[f];
    }
    als[t] = s1;
    ald[t] = s2;
}

__global__ void init_mz(int* __restrict__ mi, float* __restrict__ z) {
    int t = blockIdx.x * blockDim.x + threadIdx.x;
    if (t >= NN * HHD) return;
    mi[t] = (int)0x80000000;  // -inf key
    z[t]  = 0.0f;
}

__global__ void edge_max(const int* __restrict__ src, const int* __restrict__ dst,
                         const float* __restrict__ als, const float* __restrict__ ald,
                         int* __restrict__ mi) {
    int t = blockIdx.x * blockDim.x + threadIdx.x;
    if (t >= (EE + NN) * HHD) return;
    int e = t >> 2, hd = t & 3, s, d;
    edge_sd(e, src, dst, s, d);
    float x = als[s * HHD + hd] + ald[d * HHD + hd];
    x = x >= 0.f ? x : NEG_SLOPE * x;
    atomicMax(&mi[d * HHD + hd], fOrd(x));
}

__global__ void edge_sum(const int* __restrict__ src, const int* __restrict__ dst,
                         const float* __restrict__ als, const float* __restrict__ ald,
                         const int* __restrict__ mi, float* __restrict__ z) {
    int t = blockIdx.x * blockDim.x + threadIdx.x;
    if (t >= (EE + NN) * HHD) return;
    int e = t >> 2, hd = t & 3, s, d;
    edge_sd(e, src, dst, s, d);
    float x = als[s * HHD + hd] + ald[d * HHD + hd];
    x = x >= 0.f ? x : NEG_SLOPE * x;
    float w = expf(x - fUnord(mi[d * HHD + hd]));
    atomicAdd(&z[d * HHD + hd], w);
}

__global__ void init_bias(const float* __restrict__ b, float* __restrict__ out) {
    int t = blockIdx.x * blockDim.x + threadIdx.x;
    if (t < NN * DH) out[t] = b[t & (DH - 1)];
}

// out[dst, h, :] += alpha * hp[src, h, :]
__global__ void edge_accum(const int* __restrict__ src, const int* __restrict__ dst,
                           const float* __restrict__ als, const float* __restrict__ ald,
                           const int* __restrict__ mi, const float* __restrict__ z,
                           const float* __restrict__ hp, float* __restrict__ out) {
    int t = blockIdx.x * blockDim.x + threadIdx.x;
    if (t >= (EE + NN) * HHD) return;
    int e = t >> 2, hd = t & 3, s, d;
    edge_sd(e, src, dst, s, d);
    float x = als[s * HHD + hd] + ald[d * HHD + hd];
    x = x >= 0.f ? x : NEG_SLOPE * x;
    float alpha = expf(x - fUnord(mi[d * HHD + hd])) / z[d * HHD + hd];
    const float* sp = hp + (size_t)s * DH + hd * FHH;
    float* dp = out + (size_t)d * DH + hd * FHH;
#pragma unroll 8
    for (int f = 0; f < FHH; ++f) atomicAdd(&dp[f], sp[f] * alpha);
}

// ---------- host launch ----------
extern "C" void kernel_launch(void* const* d_in, const int* in_sizes, int n_in,
                              void* d_out, int out_size, void* d_ws, size_t ws_size,
                              hipStream_t stream) {
    (void)in_sizes; (void)n_in; (void)out_size; (void)ws_size;

    const float* x = (const float*)d_in[0];
    const int* ei  = (const int*)d_in[1];
    const int* src = ei;
    const int* dst = ei + EE;

    const size_t MAT = (size_t)NN * DH;
    float* ws  = (float*)d_ws;
    float* AGG = ws;
    float* HP  = AGG + MAT;
    float* T1  = HP + MAT;
    float* B0  = T1 + MAT;
    float* B1  = B0 + MAT;
    float* cnt = B1 + MAT;
    float* als = cnt + NN;
    float* ald = als + (size_t)NN * HHD;
    int*   mi  = (int*)(ald + (size_t)NN * HHD);
    float* z   = (float*)(mi + (size_t)NN * HHD);

    const int TPB = 256;

    // degree (same every layer)
    fillf<<<(NN + TPB - 1) / TPB, TPB, 0, stream>>>(cnt, 0.0f, NN);
    count_deg<<<(EE + TPB - 1) / TPB, TPB, 0, stream>>>(dst, cnt);

    const float* hcur = x;
    for (int l = 0; l < 3; ++l) {
        const float* Wl  = (const float*)d_in[2 + 11 * l + 0];
        const float* bl  = (const float*)d_in[2 + 11 * l + 1];
        const float* Wr  = (const float*)d_in[2 + 11 * l + 2];
        const float* Wg  = (const float*)d_in[2 + 11 * l + 3];
        const float* asr = (const float*)d_in[2 + 11 * l + 4];
        const float* ads = (const float*)d_in[2 + 11 * l + 5];
        const float* gb  = (const float*)d_in[2 + 11 * l + 6];
        const float* g1  = (const float*)d_in[2 + 11 * l + 7];
        const float* b1  = (const float*)d_in[2 + 11 * l + 8];
        const float* g2  = (const float*)d_in[2 + 11 * l + 9];
        const float* b2  = (const float*)d_in[2 + 11 * l + 10];
        float* hout = (l == 0) ? B0 : (l == 1) ? B1 : (float*)d_out;

        // SAGE: agg = segment_sum(h[src]); h1 = (agg/cnt)@Wl + bl + h@Wr
        fillf<<<(int)((MAT + TPB - 1) / TPB), TPB, 0, stream>>>(AGG, 0.0f, (int)MAT);
        scatter_add<<<(EE * 32) / TPB, TPB, 0, stream>>>(hcur, src, dst, AGG);
        gemm_wmma<true><<<NN / 32, TPB, 0, stream>>>(AGG, cnt, Wl, hcur, Wr, bl, T1);
        ln_elu<<<(NN * 32 + TPB - 1) / TPB, TPB, 0, stream>>>(T1, g1, b1, T1);

        // GAT: hp = h1 @ W
        gemm_wmma<false><<<NN / 32, TPB, 0, stream>>>(T1, nullptr, Wg, nullptr, nullptr,
                                                      nullptr, HP);
        gat_logits<<<(NN * HHD + TPB - 1) / TPB, TPB, 0, stream>>>(HP, asr, ads, als, ald);
        init_mz<<<(NN * HHD + TPB - 1) / TPB, TPB, 0, stream>>>(mi, z);
        const int ET = (EE + NN) * HHD;
        edge_max<<<(ET + TPB - 1) / TPB, TPB, 0, stream>>>(src, dst, als, ald, mi);
        edge_sum<<<(ET + TPB - 1) / TPB, TPB, 0, stream>>>(src, dst, als, ald, mi, z);
        init_bias<<<(int)((MAT + TPB - 1) / TPB), TPB, 0, stream>>>(gb, T1);
        edge_accum<<<(ET + TPB - 1) / TPB, TPB, 0, stream>>>(src, dst, als, ald, mi, z,
                                                             HP, T1);
        ln_elu<<<(NN * 32 + TPB - 1) / TPB, TPB, 0, stream>>>(T1, g2, b2, hout);

        hcur = hout;
    }
}